// VectorQuantizer_6605659701614
// MI455X (gfx1250) — compile-verified
//
#include <hip/hip_runtime.h>

// Problem constants (from reference)
#define K_CODES   8192
#define C_DIM     256
#define HW        1024          // 32*32
#define CHW       262144        // 256*1024
#define N_ROWS    16384         // 16*32*32
#define OUT_ELEMS 4194304       // 16*256*32*32
#define LDS_STRIDE 260          // 256 + 4 pad -> bank-conflict-free, float4-aligned
#define KTILES_H  256           // 512 code tiles split over 2 wave-groups

typedef __attribute__((ext_vector_type(2))) float v2f;
typedef __attribute__((ext_vector_type(4))) float f4v;
typedef __attribute__((ext_vector_type(8))) float v8f;

// ---------------------------------------------------------------------------
// Kernel 1: per-code squared norms |e_k|^2  (one wave32 per code row)
// ---------------------------------------------------------------------------
__global__ __launch_bounds__(256) void vq_norms(const float* __restrict__ emb,
                                                float* __restrict__ nrms) {
    const int lane = threadIdx.x & 31;
    const int wave = threadIdx.x >> 5;
    const int k = blockIdx.x * 8 + wave;          // 1024 blocks * 8 waves = 8192
    const float* row = emb + (size_t)k * C_DIM;
    float s = 0.f;
#pragma unroll
    for (int j = 0; j < 2; ++j) {
        f4v v = *(const f4v*)(row + lane * 8 + j * 4);
        s += v.x * v.x + v.y * v.y + v.z * v.z + v.w * v.w;
    }
#pragma unroll
    for (int off = 16; off >= 1; off >>= 1) s += __shfl_xor(s, off, 32);
    if (lane == 0) nrms[k] = s;
}

// ---------------------------------------------------------------------------
// Kernel 2: fused distance-GEMM + argmin via V_WMMA_F32_16X16X4_F32.
// 128 blocks x 8 waves over 128 z-rows. Each wave owns 32 rows (two
// accumulators) with all 128 A-fragments pinned in VGPRs, so every B-fragment
// LDS read feeds TWO WMMAs (halves LDS bandwidth per WMMA). The 512 code
// tiles are split-K: waves 0-3 scan tiles 0..255, waves 4-7 scan 256..511,
// each group double-buffering its own tile stream; partial argmins merge via
// LDS at the end (low half wins ties -> jnp.argmin semantics).
// d(n,k) = |e_k|^2 - 2*z_n.e_k   (|z_n|^2 constant per row -> dropped)
// ---------------------------------------------------------------------------
__global__ __launch_bounds__(256, 1) void vq_argmin(const float* __restrict__ z,
                                                    const float* __restrict__ emb,
                                                    const float* __restrict__ nrms,
                                                    int*   __restrict__ idxArr,
                                                    float* __restrict__ idxOutF) {
    __shared__ float zLDS[128 * LDS_STRIDE];          // 133,120 B (read once)
    __shared__ float eLDS[2][2][16 * LDS_STRIDE];     // [kgroup][buf] 4 x 16,640 B
    __shared__ float cmbV[128];                       // split-K merge buffers
    __shared__ int   cmbI[128];

    const int t    = threadIdx.x;
    const int lane = t & 31;
    const int wave = t >> 5;                          // 0..7
    const int g    = t >> 7;                          // k-group: 0 or 1
    const int tl   = t & 127;                         // thread id within group
    const int wg   = blockIdx.x;                      // 128 blocks
    const int b    = wg >> 3;                         // 8 blocks per batch image
    const int hwBase = (wg & 7) * 128;
    const size_t zBase = (size_t)b * CHW + hwBase;

    // --- stage this block's 128 z-rows into LDS (coalesced float4) ---
    for (int i = 0; i < 32; ++i) {
        const int f4idx = i * 256 + t;                // 8192 float4 total
        const int r4 = f4idx & 31;                    // 4-row group (contiguous hw)
        const int c  = f4idx >> 5;                    // channel 0..255
        f4v v = *(const f4v*)(z + zBase + (size_t)c * HW + r4 * 4);
        const int r0 = r4 * 4;
        zLDS[(r0 + 0) * LDS_STRIDE + c] = v.x;
        zLDS[(r0 + 1) * LDS_STRIDE + c] = v.y;
        zLDS[(r0 + 2) * LDS_STRIDE + c] = v.z;
        zLDS[(r0 + 3) * LDS_STRIDE + c] = v.w;
    }
    __syncthreads();

    const int rowLocal = (wave & 3) * 32;             // this wave's 32 rows
    const int halfSel  = lane >> 4;                   // 0: K={0,1}, 1: K={2,3}
    const int l15      = lane & 15;
    const int cOff     = 2 * halfSel;
    const int aRow0    = rowLocal + l15;              // A set 0: rows +0..15
    const int aRow1    = aRow0 + 16;                  // A set 1: rows +16..31

    // --- pin both A-tiles (32 rows x 256 K) in registers: 256 VGPRs ---
    v2f aF0[64], aF1[64];
#pragma unroll
    for (int i = 0; i < 64; ++i) {
        aF0[i] = *(const v2f*)(&zLDS[aRow0 * LDS_STRIDE + i * 4 + cOff]);
        aF1[i] = *(const v2f*)(&zLDS[aRow1 * LDS_STRIDE + i * 4 + cOff]);
    }

    // --- prologue: each k-group loads its tile 0 (16x256 floats) ---
#pragma unroll
    for (int j = 0; j < 8; ++j) {
        const int f4idx = j * 128 + tl;               // 1024 float4 per tile
        const int k  = f4idx >> 6;                    // 64 float4 per code row
        const int c4 = f4idx & 63;
        f4v v = *(const f4v*)(emb + (size_t)(g * 4096 + k) * C_DIM + c4 * 4);
        *(f4v*)(&eLDS[g][0][k * LDS_STRIDE + c4 * 4]) = v;
    }
    __syncthreads();

    float minv0[8], minv1[8];
    int   mini0[8], mini1[8];
#pragma unroll
    for (int j = 0; j < 8; ++j) {
        minv0[j] = __builtin_inff(); mini0[j] = 0;
        minv1[j] = __builtin_inff(); mini1[j] = 0;
    }

    for (int ktg = 0; ktg < KTILES_H; ++ktg) {
        const int kBase = (g * KTILES_H + ktg) * 16;  // this group's code tile
        const int cur   = ktg & 1;

        // --- issue next tile's global loads early (hidden behind WMMAs) ---
        f4v pf[8];
        if (ktg + 1 < KTILES_H) {
#pragma unroll
            for (int j = 0; j < 8; ++j) {
                const int f4idx = j * 128 + tl;
                const int k  = f4idx >> 6;
                const int c4 = f4idx & 63;
                pf[j] = *(const f4v*)(emb + (size_t)(kBase + 16 + k) * C_DIM + c4 * 4);
            }
        }

        // --- 32x16 S-tile: 128 fp32 WMMAs, each B fragment feeds two ---
        const float* eBuf = &eLDS[g][cur][0];
        v8f acc0 = {0.f, 0.f, 0.f, 0.f, 0.f, 0.f, 0.f, 0.f};
        v8f acc1 = {0.f, 0.f, 0.f, 0.f, 0.f, 0.f, 0.f, 0.f};
#pragma unroll
        for (int i = 0; i < 64; ++i) {
            // B fragment: lane holds B[K=4i+cOff .. +1][N=lane%16] = emb[kBase+N][K]
            v2f bb = *(const v2f*)(&eBuf[l15 * LDS_STRIDE + i * 4 + cOff]);
            acc0 = __builtin_amdgcn_wmma_f32_16x16x4_f32(
                false, aF0[i], false, bb, (short)0, acc0, false, false);
            acc1 = __builtin_amdgcn_wmma_f32_16x16x4_f32(
                false, aF1[i], false, bb, (short)0, acc1, false, false);
        }

        // --- commit prefetched tile to the alternate buffer ---
        if (ktg + 1 < KTILES_H) {
#pragma unroll
            for (int j = 0; j < 8; ++j) {
                const int f4idx = j * 128 + tl;
                const int k  = f4idx >> 6;
                const int c4 = f4idx & 63;
                *(f4v*)(&eLDS[g][cur ^ 1][k * LDS_STRIDE + c4 * 4]) = pf[j];
            }
        }

        // --- fold into running argmins: d = |e|^2 - 2*S ---
        const float nrm  = nrms[kBase + l15];
        const int   kIdx = kBase + l15;               // column N = lane%16
#pragma unroll
        for (int j = 0; j < 8; ++j) {                 // VGPR j -> row j (+8 upper half)
            const float d0 = nrm - 2.0f * acc0[j];
            if (d0 < minv0[j]) { minv0[j] = d0; mini0[j] = kIdx; }
            const float d1 = nrm - 2.0f * acc1[j];
            if (d1 < minv1[j]) { minv1[j] = d1; mini1[j] = kIdx; }
        }

        __syncthreads();                              // buffer handoff
    }

    // --- argmin reduction across the 16 lanes holding the same row ---
#pragma unroll
    for (int off = 8; off >= 1; off >>= 1) {
#pragma unroll
        for (int j = 0; j < 8; ++j) {
            float ov; int oi;
            ov = __shfl_xor(minv0[j], off, 32); oi = __shfl_xor(mini0[j], off, 32);
            if (ov < minv0[j] || (ov == minv0[j] && oi < mini0[j])) { minv0[j] = ov; mini0[j] = oi; }
            ov = __shfl_xor(minv1[j], off, 32); oi = __shfl_xor(mini1[j], off, 32);
            if (ov < minv1[j] || (ov == minv1[j] && oi < mini1[j])) { minv1[j] = ov; mini1[j] = oi; }
        }
    }

    // --- merge split-K halves: waves 4-7 publish, waves 0-3 combine ---
    const int rowW0 = rowLocal + halfSel * 8;         // local row of set 0 results
    if (g == 1 && l15 == 0) {
#pragma unroll
        for (int j = 0; j < 8; ++j) {
            cmbV[rowW0 + j]      = minv0[j];  cmbI[rowW0 + j]      = mini0[j];
            cmbV[rowW0 + 16 + j] = minv1[j];  cmbI[rowW0 + 16 + j] = mini1[j];
        }
    }
    __syncthreads();
    if (g == 0 && l15 == 0) {
        const int rowBase = wg * 128;
#pragma unroll
        for (int j = 0; j < 8; ++j) {
            // strict < : high-K half has larger indices, so ties keep low half
            float v0 = minv0[j]; int i0 = mini0[j];
            const float hv0 = cmbV[rowW0 + j]; const int hi0 = cmbI[rowW0 + j];
            if (hv0 < v0) { v0 = hv0; i0 = hi0; }
            idxArr[rowBase + rowW0 + j]  = i0;
            idxOutF[rowBase + rowW0 + j] = (float)i0;

            float v1 = minv1[j]; int i1 = mini1[j];
            const float hv1 = cmbV[rowW0 + 16 + j]; const int hi1 = cmbI[rowW0 + 16 + j];
            if (hv1 < v1) { v1 = hv1; i1 = hi1; }
            idxArr[rowBase + rowW0 + 16 + j]  = i1;
            idxOutF[rowBase + rowW0 + 16 + j] = (float)i1;
        }
    }
}

// ---------------------------------------------------------------------------
// Kernel 3: gather quantized vectors, write out in [B,C,H,W] order (coalesced),
// accumulate deterministic per-block loss partials.
// ---------------------------------------------------------------------------
__global__ __launch_bounds__(256) void vq_gather(const float* __restrict__ z,
                                                 const float* __restrict__ emb,
                                                 const int* __restrict__ idxArr,
                                                 float* __restrict__ out,
                                                 float* __restrict__ partials) {
    __shared__ float red[256];
    const int t   = threadIdx.x;
    const int blk = blockIdx.x;           // 4096 blocks: (b,c)
    const int b   = blk >> 8;
    const int c   = blk & 255;
    const size_t base = (size_t)b * CHW + (size_t)c * HW;
    float s = 0.f;
#pragma unroll
    for (int j = 0; j < 4; ++j) {
        const int hw = j * 256 + t;
        const int n  = b * HW + hw;
        const int k  = idxArr[n];
        const float q  = emb[(size_t)k * C_DIM + c];   // L2-resident gather
        const float zv = z[base + hw];
        out[base + hw] = zv + (q - zv);                // straight-through
        const float d = q - zv;
        s += d * d;
    }
    red[t] = s;
    __syncthreads();
    for (int off = 128; off >= 1; off >>= 1) {
        if (t < off) red[t] += red[t + off];
        __syncthreads();
    }
    if (t == 0) partials[blk] = red[0];
}

// ---------------------------------------------------------------------------
// Kernel 4: deterministic final loss reduction:
// loss = (1 + BETA) * mean((q-z)^2), BETA = 0.25
// ---------------------------------------------------------------------------
__global__ __launch_bounds__(256) void vq_loss(const float* __restrict__ partials,
                                               float* __restrict__ loss) {
    __shared__ float red[256];
    const int t = threadIdx.x;
    float s = 0.f;
    for (int i = t; i < 4096; i += 256) s += partials[i];
    red[t] = s;
    __syncthreads();
    for (int off = 128; off >= 1; off >>= 1) {
        if (t < off) red[t] += red[t + off];
        __syncthreads();
    }
    if (t == 0) loss[0] = red[0] * (1.25f / (float)OUT_ELEMS);
}

// ---------------------------------------------------------------------------
extern "C" void kernel_launch(void* const* d_in, const int* in_sizes, int n_in,
                              void* d_out, int out_size, void* d_ws, size_t ws_size,
                              hipStream_t stream) {
    const float* z   = (const float*)d_in[0];   // [16,256,32,32] f32
    const float* emb = (const float*)d_in[1];   // [8192,256] f32

    float* out  = (float*)d_out;                // [16,256,32,32]
    float* loss = out + OUT_ELEMS;              // scalar
    float* idxF = out + OUT_ELEMS + 1;          // [16,1024] as f32

    // workspace layout
    float* nrms     = (float*)d_ws;                                   // 8192 f32
    int*   idxArr   = (int*)((char*)d_ws + (size_t)K_CODES * 4);      // 16384 i32
    float* partials = (float*)((char*)d_ws + (size_t)(K_CODES + N_ROWS) * 4); // 4096 f32

    vq_norms <<<K_CODES / 8, 256, 0, stream>>>(emb, nrms);
    vq_argmin<<<N_ROWS / 128, 256, 0, stream>>>(z, emb, nrms, idxArr, idxF);
    vq_gather<<<16 * C_DIM, 256, 0, stream>>>(z, emb, idxArr, out, partials);
    vq_loss  <<<1, 256, 0, stream>>>(partials, loss);
}